// MeanSquaredError2_58025008169160
// MI455X (gfx1250) — compile-verified
//
#include <hip/hip_runtime.h>
#include <hip/hip_bf16.h>
#include <math.h>

#define NJ    14
#define COLS  14
#define TILE  196           // 14*14
#define NBLK  1792
#define BLOCK 256

typedef __attribute__((ext_vector_type(2))) float v2f;
typedef __attribute__((ext_vector_type(8))) float v8f;

// ---- wave32 lane-XOR butterfly via VALU permlane (no DS, no dscnt waits) ----
template <int M>
__device__ __forceinline__ int lxor_i(int x) {
  if constexpr (M == 16)
    return __builtin_amdgcn_permlanex16(x, x, 0x76543210, 0xFEDCBA98, false, false);
  else if constexpr (M == 8)
    return __builtin_amdgcn_permlane16(x, x, 0xFEDCBA98, 0x76543210, false, false);
  else if constexpr (M == 4)
    return __builtin_amdgcn_permlane16(x, x, 0x32107654, 0xBA98FEDC, false, false);
  else if constexpr (M == 2)
    return __builtin_amdgcn_permlane16(x, x, 0x54761032, 0xDCFE98BA, false, false);
  else
    return __builtin_amdgcn_permlane16(x, x, 0x67452301, 0xEFCDAB89, false, false);
}
template <int M>
__device__ __forceinline__ float lxor_f(float x) {
  return __int_as_float(lxor_i<M>(__float_as_int(x)));
}

__device__ __forceinline__ float waveRedSum(float x) {
  x += lxor_f<16>(x);
  x += lxor_f<8>(x);
  x += lxor_f<4>(x);
  x += lxor_f<2>(x);
  x += lxor_f<1>(x);
  return x;
}

template <int M>
__device__ __forceinline__ void argmaxStep(float& bmax, int& bidx) {
  float ov = lxor_f<M>(bmax);
  int   oi = lxor_i<M>(bidx);
  if (ov > bmax || (ov == bmax && oi < bidx)) { bmax = ov; bidx = oi; }
}

// One wave32 per joint. Gaussian target tile synthesized as rank-1 outer
// product with V_WMMA_F32_16X16X4_F32. Tile min/max come from precomputed
// per-column extrema of the blur matrix; all wave reductions are VALU
// permlane butterflies.
__global__ __launch_bounds__(BLOCK) void pose_loss_partial(
    const float* __restrict__ o, const float* __restrict__ h,
    const float* __restrict__ t, const float* __restrict__ v,
    float* __restrict__ partials, int nJoints)
{
  __shared__ float Klds[TILE];       // effective 1D reflect-blur matrix K[out][in]
  __shared__ float Kmn[COLS];        // per-input-column min of K
  __shared__ float Kmx[COLS];        // per-input-column max of K
  __shared__ float wred[8][3];

  if (threadIdx.x < TILE) {
    float w[9]; float wsum = 0.f;
#pragma unroll
    for (int d = 0; d < 9; ++d) { float xx = (float)(d - 4); w[d] = expf(-0.5f * xx * xx); wsum += w[d]; }
    int oo = threadIdx.x / COLS, ii = threadIdx.x % COLS;
    float acc = 0.f;
#pragma unroll
    for (int d = 0; d < 9; ++d) {
      int p = oo + d - 4;                                  // symmetric (reflect) padding
      p = (p < 0) ? (-p - 1) : ((p > COLS - 1) ? (2 * COLS - 1 - p) : p);
      if (p == ii) acc += w[d];
    }
    Klds[threadIdx.x] = acc / wsum;
  }
  __syncthreads();
  if (threadIdx.x < COLS) {
    float mn = __builtin_inff(), mx = -__builtin_inff();
#pragma unroll
    for (int r = 0; r < COLS; ++r) {
      float kv = Klds[r * COLS + threadIdx.x];
      mn = fminf(mn, kv); mx = fmaxf(mx, kv);
    }
    Kmn[threadIdx.x] = mn; Kmx[threadIdx.x] = mx;
  }
  __syncthreads();

  const int lane   = threadIdx.x & 31;
  const int wid    = threadIdx.x >> 5;
  const int hiHalf = lane >> 4;          // WMMA C/D layout: lanes 16-31 hold rows M+8
  const int N      = lane & 15;          // column index in D layout
  const int waveGlobal = blockIdx.x * 8 + wid;
  const int nWaves     = gridDim.x * 8;

  float accD1 = 0.f;    // per-lane partial of heatmap loss
  float accD2 = 0.f;    // lane-0 partial of coordinate loss
  float accVis = 0.f;   // lane-0 partial of N1

  for (int s = waveGlobal; s < nJoints; s += nWaves) {
    int b = s / NJ;
    int j = s - b * NJ;

    float tx = t[2 * s + 0], ty = t[2 * s + 1];
    float v0 = v[2 * s + 0], v1 = v[2 * s + 1];
    int  visI = ((int)v0 == 1) ? 1 : 0;

    int xi = (int)(tx * (float)COLS); xi = xi < 0 ? 0 : (xi > COLS - 1 ? COLS - 1 : xi);
    int yi = (int)(ty * (float)COLS); yi = yi < 0 ? 0 : (yi > COLS - 1 ? COLS - 1 : yi);

    // --- load h tile in WMMA C/D layout; fused per-lane argmax scan ---
    const float* hb = h + (size_t)s * TILE;
    float hv[8];
    float bmax = -__builtin_inff(); int bidx = 0x7fffffff;
#pragma unroll
    for (int r = 0; r < 8; ++r) {
      int M  = r + (hiHalf << 3);
      int Mc = M < COLS ? M : COLS - 1;
      int Nc = N < COLS ? N : COLS - 1;
      float val = hb[Mc * COLS + Nc];
      bool valid = (M < COLS) & (N < COLS);
      val = valid ? val : 0.f;
      hv[r] = val;
      if (valid & (val > bmax)) { bmax = val; bidx = M * COLS + N; }
    }
    // wave argmax (first flat index wins ties, matching jnp.argmax)
    argmaxStep<16>(bmax, bidx);
    argmaxStep<8>(bmax, bidx);
    argmaxStep<4>(bmax, bidx);
    argmaxStep<2>(bmax, bidx);
    argmaxStep<1>(bmax, bidx);

    int yC = bidx / COLS, xC = bidx - yC * COLS;
    bool cond = bmax > 0.5f;

    const float* ob = o + ((size_t)b * (2 * NJ) + j) * TILE;   // uniform gathers
    float ox = ob[yC * COLS + xC];
    float oy = ob[NJ * TILE + yC * COLS + xC];
    float x0 = cond ? (ox + (float)xC) * (1.f / COLS) : 0.f;
    float x1 = cond ? (oy + (float)yC) * (1.f / COLS) : 0.f;
    float dd0 = (x0 - tx) * v0, dd1 = (x1 - ty) * v1;
    float d2c = dd0 * dd0 + dd1 * dd1;

    // --- Gaussian target tile: D = a * b^T via V_WMMA_F32_16X16X4_F32 ---
    // A 16x4: lanes 0-15 VGPR0 hold K=0 column = a; everything else zero.
    // B 4x16: lanes 0-15 VGPR0 hold K=0 row    = b; everything else zero.
    float aval = (lane < COLS) ? Klds[lane * COLS + yi] : 0.f;
    float bval = (lane < COLS) ? Klds[lane * COLS + xi] : 0.f;
    v2f A;  A.x  = aval; A.y  = 0.f;
    v2f Bm; Bm.x = bval; Bm.y = 0.f;
    v8f C = {};
    v8f g = __builtin_amdgcn_wmma_f32_16x16x4_f32(false, A, false, Bm, (short)0, C,
                                                  false, false);

    // tile min/max factorize for a non-negative rank-1 matrix:
    // min(g) = min(a)*min(b), max(g) = max(a)*max(b) (exact: rounding is
    // monotone and the extremal pair is a realized tile element).
    float lmn = Kmn[yi] * Kmn[xi];
    float lmx = Kmx[yi] * Kmx[xi];
    float den = (lmx > lmn) ? (lmx - lmn) : 1.f;
    float invden = 1.f / den;              // one IEEE divide per joint

    // (h - tt)^2 with reference's row-0 masking for invisible joints
#pragma unroll
    for (int r = 0; r < 8; ++r) {
      int M = r + (hiHalf << 3);
      bool valid = (M < COLS) & (N < COLS);
      float tt = visI ? (g[r] - lmn) * invden : 0.f;
      float diff = hv[r] - tt;
      if ((!visI) & (M == 0)) diff = 0.f;
      accD1 += valid ? diff * diff : 0.f;
    }
    if (lane == 0) { accD2 += d2c; accVis += (float)visI; }
  }

  float wD1 = waveRedSum(accD1);
  if (lane == 0) { wred[wid][0] = wD1; wred[wid][1] = accD2; wred[wid][2] = accVis; }
  __syncthreads();
  if (threadIdx.x == 0) {
    float s1 = 0.f, s2 = 0.f, s3 = 0.f;
#pragma unroll
    for (int i = 0; i < 8; ++i) { s1 += wred[i][0]; s2 += wred[i][1]; s3 += wred[i][2]; }
    partials[blockIdx.x * 3 + 0] = s1;
    partials[blockIdx.x * 3 + 1] = s2;
    partials[blockIdx.x * 3 + 2] = s3;
  }
}

// Deterministic single-block final reduction: out = sum_d1/N1 + sum_d2/N1
__global__ __launch_bounds__(BLOCK) void pose_loss_final(
    const float* __restrict__ partials, int nblk, float* __restrict__ out)
{
  __shared__ float s1[BLOCK], s2[BLOCK], s3[BLOCK];
  int tid = threadIdx.x;
  float a = 0.f, b = 0.f, c = 0.f;
  for (int i = tid; i < nblk; i += BLOCK) {
    a += partials[3 * i + 0];
    b += partials[3 * i + 1];
    c += partials[3 * i + 2];
  }
  s1[tid] = a; s2[tid] = b; s3[tid] = c;
  __syncthreads();
  for (int m = BLOCK / 2; m > 0; m >>= 1) {
    if (tid < m) { s1[tid] += s1[tid + m]; s2[tid] += s2[tid + m]; s3[tid] += s3[tid + m]; }
    __syncthreads();
  }
  if (tid == 0) {
    float n1 = s3[0];
    out[0] = s1[0] / n1 + s2[0] / n1;
  }
}

extern "C" void kernel_launch(void* const* d_in, const int* in_sizes, int n_in,
                              void* d_out, int out_size, void* d_ws, size_t ws_size,
                              hipStream_t stream)
{
  const float* o = (const float*)d_in[0];
  const float* h = (const float*)d_in[1];
  const float* t = (const float*)d_in[2];
  const float* v = (const float*)d_in[3];
  float* out      = (float*)d_out;
  float* partials = (float*)d_ws;

  int nJoints = in_sizes[1] / TILE;        // B * NJ

  int nblk = NBLK;
  size_t need = (size_t)nblk * 3 * sizeof(float);
  if (ws_size < need) {                    // defensive: shrink grid to fit ws
    nblk = (int)(ws_size / (3 * sizeof(float)));
    if (nblk < 1) nblk = 1;
  }

  hipLaunchKernelGGL(pose_loss_partial, dim3(nblk), dim3(BLOCK), 0, stream,
                     o, h, t, v, partials, nJoints);
  hipLaunchKernelGGL(pose_loss_final, dim3(1), dim3(BLOCK), 0, stream,
                     partials, nblk, out);
}